// Memory_45964740002665
// MI455X (gfx1250) — compile-verified
//
#include <hip/hip_runtime.h>
#include <hip/hip_bf16.h>
#include <math.h>

typedef __attribute__((ext_vector_type(2))) float v2f;
typedef __attribute__((ext_vector_type(8))) float v8f;

#define WMMA_F32X4(a, b, c) \
  __builtin_amdgcn_wmma_f32_16x16x4_f32(false, (a), false, (b), (short)0, (c), false, false)

#define USE_ASYNC_LDS 1

static constexpr int N_Q   = 8192;           // queries
static constexpr int DIM   = 256;            // d
static constexpr int MSLOT = 4608;           // m
static constexpr int NCLS  = 9;
static constexpr int PART  = MSLOT / NCLS;   // 512
static constexpr int NQE   = 16384;          // Nq
static constexpr int PR    = MSLOT / (NCLS - 1); // 576
static constexpr int NBLK  = MSLOT / PR;     // 8
static constexpr int PAD   = 260;            // LDS row pitch (floats): 16B-aligned, bank-safe

// ---- monotone float<->uint encoding for atomicMax on floats ----
__device__ __forceinline__ unsigned fenc(float f) {
  unsigned u = __float_as_uint(f);
  return (u & 0x80000000u) ? ~u : (u | 0x80000000u);
}
__device__ __forceinline__ float fdec(unsigned u) {
  return (u & 0x80000000u) ? __uint_as_float(u ^ 0x80000000u)
                           : __uint_as_float(~u);
}

// ---- async DMA of 16 rows x 1024B (contiguous 16KB in global) into padded LDS ----
// Uses GLOBAL_LOAD_ASYNC_TO_LDS_B128 (ASYNCcnt); each lane copies 16B segments.
__device__ __forceinline__ void asyncCopy16Rows(const float* __restrict__ gsrc,
                                                float* ldsbuf, int tid, int nthr) {
  for (int g = tid; g < 1024; g += nthr) {  // 1024 x 16B segments
    int row = g >> 6, c16 = g & 63;
    unsigned ldsOff = (unsigned)(size_t)(ldsbuf + row * PAD + c16 * 4);
    unsigned gOff = (unsigned)(g << 4);
#if USE_ASYNC_LDS
    asm volatile("global_load_async_to_lds_b128 %0, %1, %2"
                 :: "v"(ldsOff), "v"(gOff), "s"(gsrc)
                 : "memory");
#else
    const float4 v = *(const float4*)(gsrc + (size_t)g * 4);
    *(float4*)(ldsbuf + row * PAD + c16 * 4) = v;
#endif
  }
}
__device__ __forceinline__ void asyncWait() {
#if USE_ASYNC_LDS
  asm volatile("s_wait_asynccnt 0x0" ::: "memory");
#endif
}

// ---- 256-thread block sum reduce (wave32: 8 waves) ----
__device__ __forceinline__ float blockReduceSum256(float v, float* sbuf) {
  int lane = threadIdx.x & 31, w = threadIdx.x >> 5;
  #pragma unroll
  for (int o = 16; o > 0; o >>= 1) v += __shfl_xor(v, o, 32);
  if (lane == 0) sbuf[w] = v;
  __syncthreads();
  if (w == 0) {
    float r = (lane < 8) ? sbuf[lane] : 0.0f;
    #pragma unroll
    for (int o = 4; o > 0; o >>= 1) r += __shfl_xor(r, o, 32);
    if (lane == 0) sbuf[0] = r;
  }
  __syncthreads();
  return sbuf[0];
}

// ============ kernel 1: row l2-normalize (one 256-thread block per row) ============
__global__ __launch_bounds__(256) void l2norm_rows_kernel(const float* __restrict__ in,
                                                          float* __restrict__ out) {
  __shared__ float sbuf[8];
  const int row = blockIdx.x, t = threadIdx.x;
  float v = in[(size_t)row * DIM + t];
  float ss = blockReduceSum256(v * v, sbuf);
  out[(size_t)row * DIM + t] = v / fmaxf(sqrtf(ss), 1e-12f);
}

// ============ kernel 2: zero qu + colmax ============
__global__ void init_kernel(float* __restrict__ qu, unsigned* __restrict__ colmax) {
  int idx = blockIdx.x * blockDim.x + threadIdx.x;
  if (idx < MSLOT * DIM) qu[idx] = 0.0f;
  if (idx < MSLOT) colmax[idx] = 0u;  // encodes "below every finite float"
}

// ============ kernel 3: WMMA score pass — row argmax + column max ============
// one wave per 16-query strip; score tile = qtile(16x256) @ keysTile^T(256x16)
__global__ __launch_bounds__(32) void score_kernel(const float* __restrict__ qn,
                                                   const float* __restrict__ keys,
                                                   unsigned* __restrict__ colmax,
                                                   int* __restrict__ gather) {
  __shared__ float ldsQ[16 * PAD];
  const int lane = threadIdx.x;
  const int qbase = blockIdx.x * 16;
  const int hb2 = (lane >> 4) * 2;  // K-pair base select (A/B operand layout)
  const int nlo = lane & 15;

  // async DMA of the 16x256 q tile (contiguous 16KB) into LDS
  asyncCopy16Rows(qn + (size_t)qbase * DIM, ldsQ, lane, 32);
  asyncWait();
  __syncthreads();

  float bestval[8];
  int bestcol[8];
  #pragma unroll
  for (int r = 0; r < 8; ++r) { bestval[r] = -3.4e38f; bestcol[r] = 0; }

  const v8f zero8 = {0, 0, 0, 0, 0, 0, 0, 0};
  for (int ct = 0; ct < MSLOT / 16; ++ct) {
    const int cbase = ct * 16;
    v8f acc0 = zero8, acc1 = zero8;  // dual accumulators: 2 independent WMMA chains
    const float* kp = keys + (size_t)(cbase + nlo) * DIM + hb2;
    const float* qp = ldsQ + nlo * PAD + hb2;
    #pragma unroll 4
    for (int k0 = 0; k0 < DIM; k0 += 8) {
      v2f a0, b0, a1, b1;
      a0.x = qp[k0];     a0.y = qp[k0 + 1];       // A[M=nlo][K]
      b0.x = kp[k0];     b0.y = kp[k0 + 1];       // B[K][N=nlo] = keys[col][k]
      acc0 = WMMA_F32X4(a0, b0, acc0);
      a1.x = qp[k0 + 4]; a1.y = qp[k0 + 5];
      b1.x = kp[k0 + 4]; b1.y = kp[k0 + 5];
      acc1 = WMMA_F32X4(a1, b1, acc1);
    }
    v8f acc = acc0 + acc1;
    // D layout: VGPR r -> rows (r, r+8); lanes (0-15, 16-31); N = nlo + cbase
    #pragma unroll
    for (int r = 0; r < 8; ++r) {
      float v = acc[r];
      if (v > bestval[r]) { bestval[r] = v; bestcol[r] = cbase + nlo; }
    }
    float tm = acc[0];
    #pragma unroll
    for (int r = 1; r < 8; ++r) tm = fmaxf(tm, acc[r]);
    tm = fmaxf(tm, __shfl_xor(tm, 16, 32));  // combine row halves
    if (lane < 16) atomicMax(&colmax[cbase + lane], fenc(tm));
  }

  // argmax reduce across the 16 lanes of each half; ties -> smallest col (first max)
  #pragma unroll
  for (int off = 1; off < 16; off <<= 1) {
    #pragma unroll
    for (int r = 0; r < 8; ++r) {
      float ov = __shfl_xor(bestval[r], off, 32);
      int oc = __shfl_xor(bestcol[r], off, 32);
      if (ov > bestval[r] || (ov == bestval[r] && oc < bestcol[r])) {
        bestval[r] = ov; bestcol[r] = oc;
      }
    }
  }
  if (nlo == 0) {
    #pragma unroll
    for (int r = 0; r < 8; ++r)
      gather[qbase + r + (lane >> 4) * 8] = bestcol[r];
  }
}

// ============ kernel 4: label metadata ============
__global__ void meta_kernel(const int* __restrict__ labels, int L, int* __restrict__ meta) {
  if (threadIdx.x == 0 && blockIdx.x == 0) {
    int mask = 0, last = 0;
    for (int i = 0; i < L; ++i) { int lb = labels[i]; mask |= 1 << lb; if (lb > last) last = lb; }
    int cnt = 0;
    for (int i = 0; i < L; ++i) cnt += (labels[i] == last);
    meta[0] = mask; meta[1] = last; meta[2] = cnt;
  }
}

// ============ kernel 5: scatter qu[gather[i]] += exp(score - colmax) * q[i] ============
__global__ __launch_bounds__(256) void scatter_kernel(const float* __restrict__ qn,
                                                      const float* __restrict__ keys,
                                                      const int* __restrict__ gather,
                                                      const unsigned* __restrict__ colmax,
                                                      const int* __restrict__ meta,
                                                      float* __restrict__ qu) {
  __shared__ float sbuf[8];
  const int i = blockIdx.x, t = threadIdx.x;
  const int j = gather[i];
  float qv = qn[(size_t)i * DIM + t];
  float s = blockReduceSum256(qv * keys[(size_t)j * DIM + t], sbuf);
  const int cls = j / PART;
  if ((meta[0] >> cls) & 1) {
    float w = expf(s - fdec(colmax[j]));  // s_q / max(s_q) == exp(score - colmax)
    atomicAdd(&qu[(size_t)j * DIM + t], w * qv);
  }
}

// ============ kernel 6: count-weighted update + row l2norm -> mem ============
__global__ __launch_bounds__(256) void finalize_kernel(const float* __restrict__ keys,
                                                       const float* __restrict__ qu,
                                                       const int* __restrict__ meta,
                                                       const float* __restrict__ ccnt,
                                                       float* __restrict__ memo) {
  __shared__ float sbuf[8];
  const int j = blockIdx.x, t = threadIdx.x;
  const int mask = meta[0], last = meta[1];
  const float cnt = (float)meta[2];
  const int cls = j / PART;
  const bool present = (mask >> cls) & 1;
  const bool isLast = (cls == last);
  float quv = present ? qu[(size_t)j * DIM + t] : 0.0f;
  float temp = isLast ? ccnt[last] : 1.0f;
  float denom = temp + (isLast ? cnt : 0.0f);
  float val = (temp * keys[(size_t)j * DIM + t] + quv) / denom;
  float ss = blockReduceSum256(val * val, sbuf);
  memo[(size_t)j * DIM + t] = val / fmaxf(sqrtf(ss), 1e-12f);
}

// ============ kernel 7: fused read — OUT = (E @ B^T) @ B per 16-slot chunk ============
// block = 8 waves x 16-query tiles (128 queries); grid = 2 emb * 8 blocks * 128 tiles
// Double-buffered async-to-LDS staging of the 16-slot chunk overlaps the DMA of
// chunk pc+1 with the 128 WMMAs of chunk pc.
__global__ __launch_bounds__(256) void read_kernel(const float* __restrict__ embS,
                                                   const float* __restrict__ embT,
                                                   const float* __restrict__ memo,
                                                   float* __restrict__ out) {
  __shared__ float ldsP[2][16 * PAD];    // double-buffered 16-slot chunk of B
  __shared__ float ldsS[8][16 * 18];     // per-wave S tile (16x16), padded rows
  const int tid = threadIdx.x;
  const int lane = tid & 31;
  const int w = tid >> 5;
  const int bx = blockIdx.x;
  const int tile = bx & 127;
  const int blk = (bx >> 7) & 7;
  const int e = bx >> 10;
  const float* emb = e ? embT : embS;    // [DIM][NQE] layout: E[q][k] = emb[k*NQE + q]
  const int qbase = tile * 128 + w * 16;
  const int hb2 = (lane >> 4) * 2;
  const int nlo = lane & 15;
  const float* memblk = memo + (size_t)blk * PR * DIM;
  float* Sp = &ldsS[w][0];

  v8f acc[16];
  const v8f zero8 = {0, 0, 0, 0, 0, 0, 0, 0};
  #pragma unroll
  for (int nt = 0; nt < 16; ++nt) acc[nt] = zero8;

  // prologue: DMA chunk 0 into buffer 0
  asyncCopy16Rows(memblk, &ldsP[0][0], tid, 256);

  for (int pc = 0; pc < PR / 16; ++pc) {
    const int cur = pc & 1;
    asyncWait();          // chunk pc landed in LDS (this wave's DMAs)
    __syncthreads();      // ... and every other wave's; prior buffer reads done
    if (pc + 1 < PR / 16) // prefetch chunk pc+1 into the other buffer
      asyncCopy16Rows(memblk + (size_t)(pc + 1) * 16 * DIM, &ldsP[cur ^ 1][0], tid, 256);

    // phase A: S(16x16) = E_tile(16x256) @ Pchunk^T(256x16), dual WMMA chains
    v8f s0 = zero8, s1 = zero8;
    const float* pp = &ldsP[cur][0] + nlo * PAD + hb2;
    #pragma unroll 4
    for (int k0 = 0; k0 < DIM; k0 += 8) {
      v2f a0, b0, a1, b1;
      a0.x = emb[(size_t)(k0 + hb2) * NQE + qbase + nlo];       // A[M=nlo][K]
      a0.y = emb[(size_t)(k0 + hb2 + 1) * NQE + qbase + nlo];
      b0.x = pp[k0]; b0.y = pp[k0 + 1];                         // B[K][N=nlo]=P[n][k]
      s0 = WMMA_F32X4(a0, b0, s0);
      a1.x = emb[(size_t)(k0 + 4 + hb2) * NQE + qbase + nlo];
      a1.y = emb[(size_t)(k0 + 5 + hb2) * NQE + qbase + nlo];
      b1.x = pp[k0 + 4]; b1.y = pp[k0 + 5];
      s1 = WMMA_F32X4(a1, b1, s1);
    }
    v8f s = s0 + s1;
    // spill S (D-layout) to LDS row-major so it can be re-fed as an A operand
    #pragma unroll
    for (int r = 0; r < 8; ++r)
      Sp[(r + (lane >> 4) * 8) * 18 + nlo] = s[r];
    __syncthreads();

    // phase B: OUT(16x256) += S(16x16) @ Pchunk(16x256): 16 independent acc chains
    #pragma unroll
    for (int nt = 0; nt < 16; ++nt) {
      #pragma unroll
      for (int kb = 0; kb < 4; ++kb) {
        const int kk = kb * 4 + hb2;
        v2f a; a.x = Sp[nlo * 18 + kk]; a.y = Sp[nlo * 18 + kk + 1];  // A[M=nlo][K=p]
        v2f b;
        b.x = ldsP[cur][kk * PAD + nt * 16 + nlo];        // B[K=p][N=dcol]
        b.y = ldsP[cur][(kk + 1) * PAD + nt * 16 + nlo];
        acc[nt] = WMMA_F32X4(a, b, acc[nt]);
      }
    }
  }

  // write output tile: out[e][blk][q][dcol]
  const size_t obase = ((size_t)(e * NBLK + blk) * NQE) * DIM;
  #pragma unroll
  for (int nt = 0; nt < 16; ++nt) {
    #pragma unroll
    for (int r = 0; r < 8; ++r) {
      int q = qbase + r + (lane >> 4) * 8;
      int dcol = nt * 16 + nlo;
      out[obase + (size_t)q * DIM + dcol] = acc[nt][r];
    }
  }
}

extern "C" void kernel_launch(void* const* d_in, const int* in_sizes, int n_in,
                              void* d_out, int out_size, void* d_ws, size_t ws_size,
                              hipStream_t stream) {
  (void)n_in; (void)out_size; (void)ws_size;
  const float* query  = (const float*)d_in[0];
  const float* embS   = (const float*)d_in[1];
  const float* embT   = (const float*)d_in[2];
  const float* keys   = (const float*)d_in[3];
  const float* ccnt   = (const float*)d_in[4];
  const int*   labels = (const int*)d_in[5];
  const int L = in_sizes[5];

  // workspace layout (~17.9 MB total)
  float* qn     = (float*)d_ws;                          // 8192*256
  float* memo   = qn + (size_t)N_Q * DIM;                // 4608*256
  float* qu     = memo + (size_t)MSLOT * DIM;            // 4608*256
  unsigned* cmx = (unsigned*)(qu + (size_t)MSLOT * DIM); // 4608
  int* gather   = (int*)(cmx + MSLOT);                   // 8192
  int* meta     = gather + N_Q;                          // 3 ints
  float* out    = (float*)d_out;

  l2norm_rows_kernel<<<N_Q, 256, 0, stream>>>(query, qn);
  init_kernel<<<(MSLOT * DIM + 255) / 256, 256, 0, stream>>>(qu, cmx);
  score_kernel<<<N_Q / 16, 32, 0, stream>>>(qn, keys, cmx, gather);
  meta_kernel<<<1, 32, 0, stream>>>(labels, L, meta);
  scatter_kernel<<<N_Q, 256, 0, stream>>>(qn, keys, gather, cmx, meta, qu);
  finalize_kernel<<<MSLOT, 256, 0, stream>>>(keys, qu, meta, ccnt, memo);
  read_kernel<<<2 * NBLK * (NQE / 128), 256, 0, stream>>>(embS, embT, memo, out);
}